// GATConv_62182536511730
// MI455X (gfx1250) — compile-verified
//
#include <hip/hip_runtime.h>

typedef __attribute__((ext_vector_type(2))) float v2f;
typedef __attribute__((ext_vector_type(8))) float v8f;

#define D_IN  256
#define D_OUT 64
#define ALPHA 0.2f
#define TM    32   // rows per GEMM block

// ---------------------------------------------------------------------------
// order-preserving float <-> uint encoding for exact atomic min/max
// ---------------------------------------------------------------------------
__device__ __forceinline__ unsigned fenc(float f) {
  unsigned u = __float_as_uint(f);
  return (u & 0x80000000u) ? ~u : (u | 0x80000000u);
}
__device__ __forceinline__ float fdec(unsigned e) {
  unsigned u = (e & 0x80000000u) ? (e ^ 0x80000000u) : ~e;
  return __uint_as_float(u);
}

__global__ void gat_init(unsigned* __restrict__ minmax) {
  minmax[0] = 0xFFFFFFFFu;  // encoded +inf identity for min
  minmax[1] = 0x00000000u;  // encoded -inf identity for max
}

// ---------------------------------------------------------------------------
// X_prime = X @ W via V_WMMA_F32_16X16X4_F32.
// Block = 128 threads = 4 waves; block owns TM=32 rows x 64 cols.
// A tile (32x256 f32 = 32KB) is async-staged to LDS (ASYNCcnt path) with
// coalesced b128 lanes, then each wave reads A fragments via ds_load and
// keeps TWO accumulators (two 16-row tiles sharing one B fragment) so the
// two wmma chains interleave instead of serializing on C.
// ---------------------------------------------------------------------------
__global__ __launch_bounds__(128) void gat_gemm_wmma(const float* __restrict__ X,
                                                     const float* __restrict__ W,
                                                     float* __restrict__ XP, int n) {
  __shared__ float Xs[TM * D_IN];            // 32 KB
  const int tid  = threadIdx.x;
  const int lane = tid & 31;
  const int wave = tid >> 5;                 // 0..3 -> column tile
  const int row0 = blockIdx.x * TM;
  const int n0   = wave * 16;
  const int mn15 = lane & 15;                // A: row M / B,D: col N
  const int kb   = (lane >> 4) << 1;         // lane half -> K {0,1} vs {2,3}

  // ---- async stage: 16 rounds x (128 lanes x 16B) = 32KB ----
#pragma unroll
  for (int it = 0; it < 16; ++it) {
    int fi = (it * 128 + tid) * 4;           // float index in tile, 16B aligned
    int rr = fi >> 8;                        // row within tile
    int cc = fi & (D_IN - 1);
    int gr = row0 + rr;
    if (gr >= n) gr = n - 1;                 // clamp source for tail tiles
    unsigned ldsa = (unsigned)(unsigned long long)&Xs[fi];   // low 32b = LDS offset
    unsigned long long ga = (unsigned long long)(X + (size_t)gr * D_IN + cc);
    asm volatile("global_load_async_to_lds_b128 %0, %1, off"
                 :: "v"(ldsa), "v"(ga) : "memory");
  }
  asm volatile("s_wait_asynccnt 0x0" ::: "memory");
  __syncthreads();

  const float* xr0 = &Xs[(size_t)mn15 * D_IN];          // rows 0..15 of tile
  const float* xr1 = &Xs[(size_t)(16 + mn15) * D_IN];   // rows 16..31

  v8f c0 = {}, c1 = {};
#pragma unroll 4
  for (int k = 0; k < D_IN; k += 4) {
    v2f a0f, a1f, b;
    a0f.x = xr0[k + kb];  a0f.y = xr0[k + kb + 1];      // ds_load_b64
    a1f.x = xr1[k + kb];  a1f.y = xr1[k + kb + 1];
    b.x = W[(size_t)(k + kb)     * D_OUT + n0 + mn15];  // W is L2-resident
    b.y = W[(size_t)(k + kb + 1) * D_OUT + n0 + mn15];
    c0 = __builtin_amdgcn_wmma_f32_16x16x4_f32(false, a0f, false, b,
                                               (short)0, c0, false, false);
    c1 = __builtin_amdgcn_wmma_f32_16x16x4_f32(false, a1f, false, b,
                                               (short)0, c1, false, false);
  }

  // D layout: VGPR i -> M = i (lanes 0-15) / M = 8+i (lanes 16-31), N = lane&15
  const int mbase = (lane >> 4) << 3;
  if (row0 + TM <= n) {                      // fast path: unguarded stores
    float* o = XP + (size_t)(row0 + mbase) * D_OUT + n0 + mn15;
#pragma unroll
    for (int i = 0; i < 8; ++i) {
      o[(size_t)i * D_OUT]        = c0[i];
      o[(size_t)(16 + i) * D_OUT] = c1[i];
    }
  } else {                                   // tail tile
#pragma unroll
    for (int i = 0; i < 8; ++i) {
      int r1 = row0 + mbase + i, r2 = r1 + 16;
      if (r1 < n) XP[(size_t)r1 * D_OUT + n0 + mn15] = c0[i];
      if (r2 < n) XP[(size_t)r2 * D_OUT + n0 + mn15] = c1[i];
    }
  }
}

// ---------------------------------------------------------------------------
// s0[i] = dot(X_prime[i,:], a0), s1[i] = dot(X_prime[i,:], a1)
// One wave per node: coalesced 256B row read + shfl_xor reduction.
// ---------------------------------------------------------------------------
__global__ void gat_scores(const float* __restrict__ XP,
                           const float* __restrict__ a0,
                           const float* __restrict__ a1,
                           float* __restrict__ s0, float* __restrict__ s1, int n) {
  int node = blockIdx.x * (blockDim.x >> 5) + (threadIdx.x >> 5);
  if (node >= n) return;
  const int lane = threadIdx.x & 31;
  const float* xp = XP + (size_t)node * D_OUT;
  float v0 = xp[lane], v1 = xp[lane + 32];
  float p0 = v0 * a0[lane] + v1 * a0[lane + 32];
  float p1 = v0 * a1[lane] + v1 * a1[lane + 32];
#pragma unroll
  for (int off = 16; off > 0; off >>= 1) {
    p0 += __shfl_xor(p0, off, 32);
    p1 += __shfl_xor(p1, off, 32);
  }
  if (lane == 0) { s0[node] = p0; s1[node] = p1; }
}

// ---------------------------------------------------------------------------
// att_raw[e] = leakyrelu(s0[row[e]] + s1[col[e]]); exact global min/max via
// wave shfl pre-reduce + order-independent uint atomics.
// ---------------------------------------------------------------------------
__global__ void gat_edge_att(const float* __restrict__ s0,
                             const float* __restrict__ s1,
                             const int* __restrict__ row,
                             const int* __restrict__ col,
                             float* __restrict__ att,
                             unsigned* __restrict__ minmax, int ne) {
  int e = blockIdx.x * blockDim.x + threadIdx.x;
  float v = 0.f;
  bool ok = e < ne;
  if (ok) {
    v = s0[row[e]] + s1[col[e]];
    v = v > 0.f ? v : ALPHA * v;
    att[e] = v;
  }
  unsigned umin = ok ? fenc(v) : 0xFFFFFFFFu;
  unsigned umax = ok ? umin : 0x00000000u;
#pragma unroll
  for (int off = 16; off > 0; off >>= 1) {
    unsigned o0 = __shfl_xor(umin, off, 32);
    unsigned o1 = __shfl_xor(umax, off, 32);
    umin = umin < o0 ? umin : o0;
    umax = umax > o1 ? umax : o1;
  }
  if ((threadIdx.x & 31) == 0) {
    atomicMin(&minmax[0], umin);
    atomicMax(&minmax[1], umax);
  }
}

// ---------------------------------------------------------------------------
// att[e] = exp((att[e] - mn) / (mx - mn)), in place
// ---------------------------------------------------------------------------
__global__ void gat_exp(float* __restrict__ att,
                        const unsigned* __restrict__ minmax, int ne) {
  int e = blockIdx.x * blockDim.x + threadIdx.x;
  if (e >= ne) return;
  float mn = fdec(minmax[0]);
  float mx = fdec(minmax[1]);
  float inv = 1.0f / (mx - mn);
  att[e] = expf((att[e] - mn) * inv);
}

// ---------------------------------------------------------------------------
// One wave per row (row_index sorted): binary-search the edge range, then
// deterministically accumulate h[r] and rows_sum, write h / max(sum,1e-12).
// Lane owns channels {lane, lane+32}. XP (25.6MB) stays L2-resident (192MB).
// ---------------------------------------------------------------------------
__global__ void gat_spmm(const float* __restrict__ XP,
                         const float* __restrict__ att,
                         const int* __restrict__ row,
                         const int* __restrict__ col,
                         float* __restrict__ out, int n, int ne) {
  int r = blockIdx.x * (blockDim.x >> 5) + (threadIdx.x >> 5);
  if (r >= n) return;
  const int lane = threadIdx.x & 31;

  int lo = 0, hi = ne;                       // lower_bound(row, r)
  while (lo < hi) {
    int mid = (lo + hi) >> 1;
    if (row[mid] < r) lo = mid + 1; else hi = mid;
  }
  int start = lo;
  hi = ne;                                   // lower_bound(row, r+1)
  while (lo < hi) {
    int mid = (lo + hi) >> 1;
    if (row[mid] <= r) lo = mid + 1; else hi = mid;
  }
  int end = lo;

  float acc0 = 0.f, acc1 = 0.f, s = 0.f;
  for (int e = start; e < end; ++e) {
    float a = att[e];
    int cidx = col[e];
    const float* xr = XP + (size_t)cidx * D_OUT;
    acc0 += a * xr[lane];
    acc1 += a * xr[lane + 32];
    s += a;
  }
  float denom = fmaxf(s, 1e-12f);
  out[(size_t)r * D_OUT + lane]      = acc0 / denom;
  out[(size_t)r * D_OUT + lane + 32] = acc1 / denom;
}

// ---------------------------------------------------------------------------
extern "C" void kernel_launch(void* const* d_in, const int* in_sizes, int n_in,
                              void* d_out, int out_size, void* d_ws, size_t ws_size,
                              hipStream_t stream) {
  const float* X  = (const float*)d_in[0];
  const float* W  = (const float*)d_in[1];
  const float* a0 = (const float*)d_in[2];
  const float* a1 = (const float*)d_in[3];
  const int* row  = (const int*)d_in[4];
  const int* col  = (const int*)d_in[5];
  float* out = (float*)d_out;

  const int n  = in_sizes[0] / D_IN;   // 100000 nodes
  const int ne = in_sizes[4];          // 1600000 edges

  // workspace (floats): XP[n*64] | s0[n] | s1[n] | att[ne] | minmax(2 u32)
  float* xp  = (float*)d_ws;
  float* s0  = xp + (size_t)n * D_OUT;
  float* s1  = s0 + n;
  float* att = s1 + n;
  unsigned* minmax = (unsigned*)(att + ne);

  gat_init<<<1, 1, 0, stream>>>(minmax);
  gat_gemm_wmma<<<(n + TM - 1) / TM, 128, 0, stream>>>(X, W, xp, n);
  gat_scores<<<(n + 7) / 8, 256, 0, stream>>>(xp, a0, a1, s0, s1, n);
  gat_edge_att<<<(ne + 255) / 256, 256, 0, stream>>>(s0, s1, row, col, att, minmax, ne);
  gat_exp<<<(ne + 255) / 256, 256, 0, stream>>>(att, minmax, ne);
  gat_spmm<<<(n + 7) / 8, 256, 0, stream>>>(xp, att, row, col, out, n, ne);
}